// SRULayer_85143431676052
// MI455X (gfx1250) — compile-verified
//
#include <hip/hip_runtime.h>
#include <hip/hip_bf16.h>
#include <math.h>

typedef float v2f __attribute__((ext_vector_type(2)));
typedef float v8f __attribute__((ext_vector_type(8)));

#define L_DIM  2048
#define B_DIM  8
#define D_DIM  1024
#define M_DIM  (L_DIM * B_DIM)   // 16384 rows
#define N3_DIM (3 * D_DIM)       // 3072 cols (u|f|r)
#define K_DIM  D_DIM             // 1024

__device__ __forceinline__ float fast_sigmoid(float x) {
    // v_exp_f32 + v_rcp_f32 (avoid IEEE div_scale/div_fixup expansion)
    return __builtin_amdgcn_rcpf(1.0f + __expf(-x));
}
__device__ __forceinline__ float fast_tanh(float x) {
#if __has_builtin(__builtin_amdgcn_tanhf)
    return __builtin_amdgcn_tanhf(x);   // native v_tanh_f32 on gfx1250
#else
    // tanh(x) = 1 - 2/(exp(2x)+1)
    return 1.0f - 2.0f * __builtin_amdgcn_rcpf(__expf(2.0f * x) + 1.0f);
#endif
}
__device__ __forceinline__ float gelu_erf_f(float x) {
    // exact erf gelu (torch F.gelu default)
    return 0.5f * x * (1.0f + erff(x * 0.70710678118654752440f));
}

// ---------------------------------------------------------------------------
// GEMM: ufr[M, 3D] = X[M, K] @ W[K, 3D], sigmoid fused for cols >= D (f,r)
// fp32 WMMA 16x16x4. Block = 8 waves (2 M x 4 N), block tile 128x256,
// wave tile 64x64 = 4x4 WMMA fragments (128 accum VGPRs).
// The u/f/r column regions (width 1024 = 4 block tiles) are exact multiples
// of the 256-wide block tile, so the sigmoid gate is block-uniform.
// ---------------------------------------------------------------------------
__global__ __launch_bounds__(256) void sru_gemm_wmma_f32(
    const float* __restrict__ X, const float* __restrict__ W,
    float* __restrict__ UFR) {
    const int lane = threadIdx.x & 31;
    const int wave = threadIdx.x >> 5;
    const int half = lane >> 4;   // k-pair select: {k,k+1} vs {k+2,k+3}
    const int ln   = lane & 15;
    const int wm = wave & 1;
    const int wn = wave >> 1;
    const int m0 = blockIdx.y * 128 + wm * 64;
    const int n0 = blockIdx.x * 256 + wn * 64;
    const bool gate = (blockIdx.x >= 4);  // scalar: whole block is f/r region

    v8f acc[4][4];
#pragma unroll
    for (int i = 0; i < 4; ++i)
#pragma unroll
        for (int j = 0; j < 4; ++j) {
            v8f z = {0.f, 0.f, 0.f, 0.f, 0.f, 0.f, 0.f, 0.f};
            acc[i][j] = z;
        }

    for (int k = 0; k < K_DIM; k += 4) {
        // speculative prefetch of next K-chunk (global_prefetch_b8)
        const int kn = (k + 4 < K_DIM) ? (k + 4) : k;
        __builtin_prefetch(X + (size_t)(m0 + ln) * K_DIM + kn + 2 * half, 0, 1);
        __builtin_prefetch(W + (size_t)(kn + 2 * half) * N3_DIM + n0 + ln, 0, 1);

        // A fragments (16x4 f32): lane holds k+2*half, k+2*half+1 of row m0+i*16+ln
        v2f a[4];
#pragma unroll
        for (int i = 0; i < 4; ++i) {
            const float* ap =
                X + (size_t)(m0 + i * 16 + ln) * K_DIM + (k + 2 * half);
            a[i] = *(const v2f*)ap;  // 8B aligned (k even, rows 4KB-aligned)
        }
        // B fragments (4x16 f32): VGPR0 = row k (lanes0-15) / k+2 (lanes16-31)
        v2f b[4];
#pragma unroll
        for (int j = 0; j < 4; ++j) {
            const float* bp =
                W + (size_t)(k + 2 * half) * N3_DIM + (n0 + j * 16 + ln);
            b[j].x = bp[0];
            b[j].y = bp[N3_DIM];
        }
#pragma unroll
        for (int i = 0; i < 4; ++i)
#pragma unroll
            for (int j = 0; j < 4; ++j)
                acc[i][j] = __builtin_amdgcn_wmma_f32_16x16x4_f32(
                    false, a[i], false, b[j], (short)0, acc[i][j], false, false);
    }

    // Epilogue: C/D layout — VGPR v holds rows M=v (lanes 0-15), M=v+8 (16-31)
#pragma unroll
    for (int i = 0; i < 4; ++i) {
#pragma unroll
        for (int j = 0; j < 4; ++j) {
            const int col = n0 + j * 16 + ln;
#pragma unroll
            for (int v = 0; v < 8; ++v) {
                const int row = m0 + i * 16 + v + 8 * half;
                float val = acc[i][j][v];
                if (gate) val = fast_sigmoid(val);  // block-uniform branch
                UFR[(size_t)row * N3_DIM + col] = val;
            }
        }
    }
}

// ---------------------------------------------------------------------------
// SRU scan over L, fused gelu. One thread per (b,d) channel; writes g=gelu(h)
// into d_out and last_c into the output tail. Coalesced along d.
// ---------------------------------------------------------------------------
__global__ __launch_bounds__(256) void sru_scan_gelu(
    const float* __restrict__ X, const float* __restrict__ C0,
    const float* __restrict__ UFR, float* __restrict__ OUT,
    float* __restrict__ LASTC) {
    const int ch = blockIdx.x * blockDim.x + threadIdx.x;  // 0..B*D-1
    const int b = ch / D_DIM;
    const int d = ch - b * D_DIM;
    float c = C0[ch];
    for (int t = 0; t < L_DIM; ++t) {
        const size_t row = (size_t)t * B_DIM + b;
        const size_t ub = row * N3_DIM;
        const float u = UFR[ub + d];
        const float f = UFR[ub + D_DIM + d];   // already sigmoid'ed
        const float r = UFR[ub + 2 * D_DIM + d];
        const float x = X[row * D_DIM + d];
        c = f * c + (1.0f - f) * u;
        const float h = r * fast_tanh(c) + (1.0f - r) * x;
        OUT[row * D_DIM + d] = gelu_erf_f(h);
    }
    LASTC[ch] = c;
}

// ---------------------------------------------------------------------------
// In-place LayerNorm over D per row. 256 threads x 4 elems; wave32 shuffle
// reduction + tiny LDS cross-wave combine. Population variance, eps=1e-5.
// ---------------------------------------------------------------------------
__global__ __launch_bounds__(256) void sru_layernorm(
    float* __restrict__ OUT, const float* __restrict__ GAMMA,
    const float* __restrict__ BETA) {
    float* p = OUT + (size_t)blockIdx.x * D_DIM;
    float vals[4];
    float s = 0.f, s2 = 0.f;
#pragma unroll
    for (int i = 0; i < 4; ++i) {
        vals[i] = p[threadIdx.x + i * 256];
        s += vals[i];
        s2 += vals[i] * vals[i];
    }
#pragma unroll
    for (int off = 16; off > 0; off >>= 1) {
        s += __shfl_down(s, off, 32);
        s2 += __shfl_down(s2, off, 32);
    }
    __shared__ float ls[8], ls2[8];
    const int lane = threadIdx.x & 31, w = threadIdx.x >> 5;
    if (lane == 0) { ls[w] = s; ls2[w] = s2; }
    __syncthreads();
    if (w == 0) {
        s = (lane < 8) ? ls[lane] : 0.f;
        s2 = (lane < 8) ? ls2[lane] : 0.f;
#pragma unroll
        for (int off = 4; off > 0; off >>= 1) {
            s += __shfl_down(s, off, 32);
            s2 += __shfl_down(s2, off, 32);
        }
        if (lane == 0) { ls[0] = s; ls2[0] = s2; }
    }
    __syncthreads();
    const float mu = ls[0] * (1.0f / D_DIM);
    const float var = ls2[0] * (1.0f / D_DIM) - mu * mu;
    const float inv = rsqrtf(var + 1e-5f);
#pragma unroll
    for (int i = 0; i < 4; ++i) {
        const int d = threadIdx.x + i * 256;
        p[d] = (vals[i] - mu) * inv * GAMMA[d] + BETA[d];
    }
}

extern "C" void kernel_launch(void* const* d_in, const int* in_sizes, int n_in,
                              void* d_out, int out_size, void* d_ws,
                              size_t ws_size, hipStream_t stream) {
    (void)in_sizes; (void)n_in; (void)out_size; (void)ws_size;
    const float* x     = (const float*)d_in[0];  // [L,B,D]
    const float* c0    = (const float*)d_in[1];  // [B,D]
    const float* W     = (const float*)d_in[2];  // [D,3D]
    const float* gamma = (const float*)d_in[3];  // [D]
    const float* beta  = (const float*)d_in[4];  // [D]

    float* out   = (float*)d_out;                 // [L,B,D]
    float* lastc = out + (size_t)M_DIM * D_DIM;   // [B,D] tail
    float* ufr   = (float*)d_ws;                  // [M,3D] = 192 MB scratch

    dim3 gemm_grid(N3_DIM / 256, M_DIM / 128);    // 12 x 128 = 1536 blocks
    sru_gemm_wmma_f32<<<gemm_grid, 256, 0, stream>>>(x, W, ufr);
    sru_scan_gelu<<<(B_DIM * D_DIM) / 256, 256, 0, stream>>>(x, c0, ufr, out,
                                                             lastc);
    sru_layernorm<<<M_DIM, 256, 0, stream>>>(out, gamma, beta);
}